// Sinkhorn_sparse_39573828665618
// MI455X (gfx1250) — compile-verified
//
#include <hip/hip_runtime.h>
#include <hip/hip_bf16.h>
#include <math.h>

typedef __attribute__((ext_vector_type(2))) float v2f;
typedef __attribute__((ext_vector_type(8))) float v8f;

#define NROW 8192
#define NCOL 16384
#define TEMPF 50.0f
#define TOPK 10
#define COL_CHUNKS 8
#define ROWS_PER_CHUNK (NROW / COL_CHUNKS) // 1024
#define BLK 256

// ---------------------------------------------------------------------------
// Kernel 1: S0 = exp(50*sims); u = 1 / rowsum(S0)     (iteration k=0)
// One block per row; float4 streaming; fixed-order LDS tree reduction.
// ---------------------------------------------------------------------------
__global__ void k_exp_rowsum(const float* __restrict__ sims,
                             float* __restrict__ S0,
                             float* __restrict__ u) {
    __shared__ float red[BLK];
    const int row = blockIdx.x;
    const int tid = threadIdx.x;
    const float4* in4  = (const float4*)(sims + (size_t)row * NCOL);
    float4*       out4 = (float4*)(S0 + (size_t)row * NCOL);
    float acc = 0.0f;
    for (int k = tid; k < NCOL / 4; k += BLK) {
        if (k + BLK < NCOL / 4) __builtin_prefetch((const void*)&in4[k + BLK], 0, 1);
        float4 x = in4[k];
        float4 e;
        e.x = expf(TEMPF * x.x);
        e.y = expf(TEMPF * x.y);
        e.z = expf(TEMPF * x.z);
        e.w = expf(TEMPF * x.w);
        out4[k] = e;
        acc += (e.x + e.y) + (e.z + e.w);
    }
    red[tid] = acc;
    __syncthreads();
    for (int s = BLK / 2; s > 0; s >>= 1) {
        if (tid < s) red[tid] += red[tid + s];
        __syncthreads();
    }
    if (tid == 0) u[row] = 1.0f / red[0];
}

// ---------------------------------------------------------------------------
// Kernel 2: column pass via fp32 WMMA.  partial[chunk][j] = sum_i u_i*S0_ij
// over a 1024-row chunk.  Per wave: one 16-column tile.
//   A (16x4 f32): row 0 = u[i..i+3], rows 1..15 zero.
//     Branch-free construction: u[i..i+3] is wave-uniform -> SMEM (s_load),
//     lane select via v_cndmask, row-0 masking via multiply by sel in {0,1}.
//   B (4x16 f32): B[k][n] = S0[i+k][j0+n]
//     lanes 0-15: K0/K1 rows (contiguous 16-float segments); lanes16-31: K2/K3
//   D[0][n] (lanes 0-15, VGPR0) accumulates the weighted column sums.
// ---------------------------------------------------------------------------
__global__ void k_colpass_wmma(const float* __restrict__ S0,
                               const float* __restrict__ u,
                               float* __restrict__ partial) {
    const int lane  = threadIdx.x & 31;
    const int wave  = threadIdx.x >> 5;
    const int n     = lane & 15;
    const int khalf = lane >> 4;               // 0: K=0,1  1: K=2,3
    const int j0    = (blockIdx.x * 8 + wave) * 16;
    const int iBase = blockIdx.y * ROWS_PER_CHUNK;
    const float sel = (n == 0) ? 1.0f : 0.0f;  // only M=0 row of A is live

    // per-lane B base: row (iBase + 2*khalf), column j0+n
    const float* __restrict__ p =
        S0 + (size_t)(iBase + 2 * khalf) * NCOL + (size_t)(j0 + n);
    const float* __restrict__ up = u + iBase;

    v8f c = {0.f, 0.f, 0.f, 0.f, 0.f, 0.f, 0.f, 0.f};
    for (int i = 0; i < ROWS_PER_CHUNK; i += 4) {
        // wave-uniform scalar loads of the 4 u values for this K-step
        const float u0 = up[i + 0];
        const float u1 = up[i + 1];
        const float u2 = up[i + 2];
        const float u3 = up[i + 3];
        v2f a;
        a.x = sel * (khalf ? u2 : u0);
        a.y = sel * (khalf ? u3 : u1);
        v2f b;
        b.x = p[0];
        b.y = p[NCOL];
        c = __builtin_amdgcn_wmma_f32_16x16x4_f32(false, a, false, b,
                                                  (short)0, c, false, false);
        p += (size_t)4 * NCOL;
    }
    if (lane < 16)
        partial[(size_t)blockIdx.y * NCOL + j0 + n] = c[0];
}

// ---------------------------------------------------------------------------
// Kernel 3: v[j] = 1 / sum_chunks partial[chunk][j]   (fixed order -> determ.)
// ---------------------------------------------------------------------------
__global__ void k_vreduce(const float* __restrict__ partial,
                          float* __restrict__ v) {
    const int j = blockIdx.x * BLK + threadIdx.x;
    float s = 0.0f;
    for (int c = 0; c < COL_CHUNKS; ++c)
        s += partial[(size_t)c * NCOL + j];
    v[j] = 1.0f / s;
}

// ---------------------------------------------------------------------------
// Kernel 4: row pass.  u[i] = 1 / sum_j S0_ij * v_j   (float4 streaming)
// ---------------------------------------------------------------------------
__global__ void k_rowpass(const float* __restrict__ S0,
                          const float* __restrict__ v,
                          float* __restrict__ u) {
    __shared__ float red[BLK];
    const int row = blockIdx.x;
    const int tid = threadIdx.x;
    const float4* s4 = (const float4*)(S0 + (size_t)row * NCOL);
    const float4* v4 = (const float4*)v;
    float acc = 0.0f;
    for (int k = tid; k < NCOL / 4; k += BLK) {
        if (k + BLK < NCOL / 4) __builtin_prefetch((const void*)&s4[k + BLK], 0, 1);
        float4 a = s4[k];
        float4 b = v4[k];
        acc += a.x * b.x + a.y * b.y + a.z * b.z + a.w * b.w;
    }
    red[tid] = acc;
    __syncthreads();
    for (int s = BLK / 2; s > 0; s >>= 1) {
        if (tid < s) red[tid] += red[tid + s];
        __syncthreads();
    }
    if (tid == 0) u[row] = 1.0f / red[0];
}

// ---------------------------------------------------------------------------
// Kernel 5: finalize.  s_ij = u_i * S0_ij * v_j (in place), per-row argmax +
// top-10 column indices.  Per-thread sorted insertion, LDS merge tree.
// ---------------------------------------------------------------------------
#define TRY_INSERT(val, idx)                                                   \
    do {                                                                       \
        float _v = (val);                                                      \
        int _i = (idx);                                                        \
        if (_v > tv[TOPK - 1] || (_v == tv[TOPK - 1] && _i < ti[TOPK - 1])) {  \
            tv[TOPK - 1] = _v;                                                 \
            ti[TOPK - 1] = _i;                                                 \
            _Pragma("unroll") for (int m = TOPK - 1; m > 0; --m) {             \
                bool sw = (tv[m] > tv[m - 1]) ||                               \
                          ((tv[m] == tv[m - 1]) && (ti[m] < ti[m - 1]));       \
                float fv = sw ? tv[m - 1] : tv[m];                             \
                int fi = sw ? ti[m - 1] : ti[m];                               \
                tv[m - 1] = sw ? tv[m] : tv[m - 1];                            \
                ti[m - 1] = sw ? ti[m] : ti[m - 1];                            \
                tv[m] = fv;                                                    \
                ti[m] = fi;                                                    \
            }                                                                  \
        }                                                                      \
    } while (0)

__global__ void k_finalize(float* __restrict__ s,   // holds S0, becomes output
                           const float* __restrict__ u,
                           const float* __restrict__ v,
                           float* __restrict__ outIdx,   // 2*NROW
                           float* __restrict__ outVal,   // NROW
                           float* __restrict__ outCols)  // NROW*TOPK
{
    __shared__ float sv[BLK * TOPK];
    __shared__ int   si[BLK * TOPK];
    const int row = blockIdx.x;
    const int tid = threadIdx.x;
    const float ui = u[row];
    float4*       s4 = (float4*)(s + (size_t)row * NCOL);
    const float4* v4 = (const float4*)v;

    float tv[TOPK];
    int   ti[TOPK];
#pragma unroll
    for (int m = 0; m < TOPK; ++m) { tv[m] = -3.0e38f; ti[m] = NCOL; }

    for (int k = tid; k < NCOL / 4; k += BLK) {
        if (k + BLK < NCOL / 4) __builtin_prefetch((const void*)&s4[k + BLK], 0, 1);
        float4 a = s4[k];
        float4 b = v4[k];
        float t0 = a.x * b.x, t1 = a.y * b.y, t2 = a.z * b.z, t3 = a.w * b.w;
        float4 o;
        o.x = ui * t0; o.y = ui * t1; o.z = ui * t2; o.w = ui * t3;
        s4[k] = o;
        const int jb = k * 4;
        TRY_INSERT(t0, jb + 0);
        TRY_INSERT(t1, jb + 1);
        TRY_INSERT(t2, jb + 2);
        TRY_INSERT(t3, jb + 3);
    }

#pragma unroll
    for (int m = 0; m < TOPK; ++m) {
        sv[tid * TOPK + m] = tv[m];
        si[tid * TOPK + m] = ti[m];
    }
    __syncthreads();

    for (int stride = BLK / 2; stride > 0; stride >>= 1) {
        if (tid < stride) {
            float* aV = sv + tid * TOPK;
            int*   aI = si + tid * TOPK;
            float* bV = sv + (tid + stride) * TOPK;
            int*   bI = si + (tid + stride) * TOPK;
            float mv[TOPK];
            int   mi[TOPK];
            int pa = 0, pb = 0;
#pragma unroll
            for (int m = 0; m < TOPK; ++m) {
                float av = aV[pa], bv = bV[pb];
                int   ai = aI[pa], bi = bI[pb];
                bool takeA = (av > bv) || (av == bv && ai <= bi);
                mv[m] = takeA ? av : bv;
                mi[m] = takeA ? ai : bi;
                pa += takeA ? 1 : 0;
                pb += takeA ? 0 : 1;
            }
#pragma unroll
            for (int m = 0; m < TOPK; ++m) { aV[m] = mv[m]; aI[m] = mi[m]; }
        }
        __syncthreads();
    }

    if (tid == 0) {
        outIdx[row]        = (float)row;    // indices[0][row]
        outIdx[NROW + row] = (float)si[0];  // indices[1][row] = argmax col
        outVal[row] = 1.0f;
        for (int m = 0; m < TOPK; ++m)
            outCols[(size_t)row * TOPK + m] = (float)si[m];
    }
}

// ---------------------------------------------------------------------------
// kernel_launch: Sinkhorn as diagonal-scaling matvec iterations.
//   k=0: u = 1/rowsum(S0)                  (k_exp_rowsum, fused with exp)
//   k=1,3,5,7,9: v = 1/(S0^T u)            (WMMA col pass + reduce)
//   k=2,4,6,8:   u = 1/(S0 v)              (row pass)
//   finalize: s = diag(u) S0 diag(v), argmax, top-10
// ---------------------------------------------------------------------------
extern "C" void kernel_launch(void* const* d_in, const int* in_sizes, int n_in,
                              void* d_out, int out_size, void* d_ws, size_t ws_size,
                              hipStream_t stream) {
    const float* sims = (const float*)d_in[0];
    float* out = (float*)d_out;

    float* S0      = out;                            // s region, doubles as S0 scratch
    float* outIdx  = out + (size_t)NROW * NCOL;      // 2*NROW
    float* outVal  = outIdx + 2 * NROW;              // NROW
    float* outCols = outVal + NROW;                  // NROW*TOPK

    float* ws      = (float*)d_ws;
    float* u       = ws;                             // NROW
    float* v       = ws + NROW;                      // NCOL
    float* partial = v + NCOL;                       // COL_CHUNKS * NCOL

    // k = 0: exp + row normalize factor
    k_exp_rowsum<<<NROW, BLK, 0, stream>>>(sims, S0, u);

    for (int it = 0; it < 5; ++it) {
        // odd k (1,3,5,7,9): column pass (WMMA) -> v
        k_colpass_wmma<<<dim3(NCOL / 128, COL_CHUNKS), BLK, 0, stream>>>(S0, u, partial);
        k_vreduce<<<NCOL / BLK, BLK, 0, stream>>>(partial, v);
        // even k (2,4,6,8): row pass -> u
        if (it < 4)
            k_rowpass<<<NROW, BLK, 0, stream>>>(S0, v, u);
    }

    k_finalize<<<NROW, BLK, 0, stream>>>(S0, u, v, outIdx, outVal, outCols);
}